// G2GEncoder_36034775613533
// MI455X (gfx1250) — compile-verified
//
#include <hip/hip_runtime.h>

// ---- problem constants (fixed by the reference) ----
#define NG   100000
#define EG   400000
#define ELG  1200000
#define NT   50000
#define ET   100000
#define ELGT 300000

typedef __attribute__((ext_vector_type(2))) float v2f;
typedef __attribute__((ext_vector_type(8))) float v8f;

// -------------------- WMMA f32 16x16x4 helpers --------------------
__device__ __forceinline__ v8f wmma_f32x4(v2f a, v2f b, v8f c) {
  // D = A(16x4) * B(4x16) + C(16x16), full fp32
  return __builtin_amdgcn_wmma_f32_16x16x4_f32(false, a, false, b, (short)0, c,
                                               false, false);
}

// C(16x16) tile at column col0: acc += A[16xK](LDS, stride lda) * B[Kx128](global)
__device__ __forceinline__ v8f gemm_tile(const float* __restrict__ As, int lda,
                                         const float* __restrict__ Bg, int K,
                                         int col0, v8f acc) {
  const int lane = threadIdx.x & 31;
  const int l15  = lane & 15;
  const int half = lane >> 4;
  const float* ap = As + l15 * lda + 2 * half;              // A: (m=l15, k=2*half+v)
  const float* bp = Bg + (2 * half) * 128 + col0 + l15;     // B: (k=2*half+v, n=l15)
#pragma unroll 4
  for (int k = 0; k < K; k += 4) {
    v2f a; a.x = ap[k];        a.y = ap[k + 1];
    v2f b; b.x = bp[k * 128];  b.y = bp[(k + 1) * 128];
    acc = wmma_f32x4(a, b, acc);
  }
  return acc;
}

__device__ __forceinline__ v8f load_c(const float* __restrict__ C, int col0) {
  const int lane = threadIdx.x & 31;
  const int l15  = lane & 15;
  const int half = lane >> 4;
  v8f c;
#pragma unroll
  for (int r = 0; r < 8; ++r)
    c[r] = C[(size_t)(r + 8 * half) * 128 + col0 + l15];
  return c;
}

__device__ __forceinline__ void store_c(float* __restrict__ D, int col0, v8f c,
                                        bool relu) {
  const int lane = threadIdx.x & 31;
  const int l15  = lane & 15;
  const int half = lane >> 4;
#pragma unroll
  for (int r = 0; r < 8; ++r) {
    float v = c[r];
    if (relu) v = fmaxf(v, 0.0f);
    D[(size_t)(r + 8 * half) * 128 + col0 + l15] = v;
  }
}

__device__ __forceinline__ float sigmoidf_(float x) {
  return 1.0f / (1.0f + __expf(-x));
}

// -------------------- kernels --------------------

// base_G = G_f[G_src] @ W1 + G_ef @ W2   (K=64 each, N=128)
__global__ void k_base_G(const float* __restrict__ Gf, const float* __restrict__ Gef,
                         const int* __restrict__ Gsrc, const float* __restrict__ W1,
                         const float* __restrict__ W2, float* __restrict__ base) {
  __shared__ float Af[16 * 64];
  __shared__ float Ae[16 * 64];
  const int row0 = blockIdx.x * 16;
  const int tid  = threadIdx.x;
  {
    const int r  = tid >> 4;            // 0..15
    const int c4 = (tid & 15) * 4;      // 0..60
    const int s  = Gsrc[row0 + r];
    *(float4*)(Af + r * 64 + c4) = *(const float4*)(Gf  + (size_t)s * 64 + c4);
    *(float4*)(Ae + r * 64 + c4) = *(const float4*)(Gef + (size_t)(row0 + r) * 64 + c4);
  }
  __syncthreads();
  const int col0 = (tid >> 5) * 16;
  v8f acc = {};
  acc = gemm_tile(Af, 64, W1, 64, col0, acc);
  acc = gemm_tile(Ae, 64, W2, 64, col0, acc);
  store_c(base + (size_t)row0 * 128, col0, acc, false);
}

// msg = relu(base + sum @ W3)
__global__ void k_msg_update(const float* __restrict__ sum,
                             const float* __restrict__ base,
                             const float* __restrict__ W3,
                             float* __restrict__ msg) {
  __shared__ float As[16 * 128];
  const int row0 = blockIdx.x * 16;
  const int tid  = threadIdx.x;
  {
    const int r = tid >> 4;
    const int c = (tid & 15) * 8;
    *(float4*)(As + r * 128 + c)     = *(const float4*)(sum + (size_t)(row0 + r) * 128 + c);
    *(float4*)(As + r * 128 + c + 4) = *(const float4*)(sum + (size_t)(row0 + r) * 128 + c + 4);
  }
  __syncthreads();
  const int col0 = (tid >> 5) * 16;
  v8f acc = load_c(base + (size_t)row0 * 128, col0);
  acc = gemm_tile(As, 128, W3, 128, col0, acc);
  store_c(msg + (size_t)row0 * 128, col0, acc, true);
}

// out = relu(F @ U1 + S @ U2); F: [rows x KF], S: [rows x 128]
template <int KF>
__global__ void k_readout(const float* __restrict__ F, const float* __restrict__ S,
                          const float* __restrict__ U1, const float* __restrict__ U2,
                          float* __restrict__ out) {
  __shared__ float Af[16 * KF];
  __shared__ float As[16 * 128];
  const int row0 = blockIdx.x * 16;
  const int tid  = threadIdx.x;
  for (int i = tid; i < 16 * (KF / 4); i += 256) {
    const int r = i / (KF / 4);
    const int c = (i % (KF / 4)) * 4;
    *(float4*)(Af + r * KF + c) = *(const float4*)(F + (size_t)(row0 + r) * KF + c);
  }
  for (int i = tid; i < 16 * 32; i += 256) {
    const int r = i >> 5;
    const int c = (i & 31) * 4;
    *(float4*)(As + r * 128 + c) = *(const float4*)(S + (size_t)(row0 + r) * 128 + c);
  }
  __syncthreads();
  const int col0 = (tid >> 5) * 16;
  v8f acc = {};
  acc = gemm_tile(Af, KF, U1, KF, col0, acc);
  acc = gemm_tile(As, 128, U2, 128, col0, acc);
  store_c(out + (size_t)row0 * 128, col0, acc, true);
}

// xz/xr/xh = f_T[T_src] @ {Wz,Wr,Wh}
__global__ void k_xzrh(const float* __restrict__ fT, const int* __restrict__ Tsrc,
                       const float* __restrict__ Wz, const float* __restrict__ Wr,
                       const float* __restrict__ Wh, float* __restrict__ xz,
                       float* __restrict__ xr, float* __restrict__ xh) {
  __shared__ float As[16 * 128];
  const int row0 = blockIdx.x * 16;
  const int tid  = threadIdx.x;
  {
    const int r = tid >> 4;
    const int c = (tid & 15) * 8;
    const int s = Tsrc[row0 + r];
    *(float4*)(As + r * 128 + c)     = *(const float4*)(fT + (size_t)s * 128 + c);
    *(float4*)(As + r * 128 + c + 4) = *(const float4*)(fT + (size_t)s * 128 + c + 4);
  }
  __syncthreads();
  const int col0 = (tid >> 5) * 16;
  v8f az = {}, ar = {}, ah = {};
  az = gemm_tile(As, 128, Wz, 128, col0, az);
  ar = gemm_tile(As, 128, Wr, 128, col0, ar);
  ah = gemm_tile(As, 128, Wh, 128, col0, ah);
  store_c(xz + (size_t)row0 * 128, col0, az, false);
  store_c(xr + (size_t)row0 * 128, col0, ar, false);
  store_c(xh + (size_t)row0 * 128, col0, ah, false);
}

// one GRU step: z=sig(xz+s@Uz), r=sig(xr+s@Ur), h=tanh(xh+(r*s)@Uh), msg=(1-z)s+zh
__global__ void k_gru(const float* __restrict__ s_in, const float* __restrict__ xz,
                      const float* __restrict__ xr, const float* __restrict__ xh,
                      const float* __restrict__ Uz, const float* __restrict__ Ur,
                      const float* __restrict__ Uh, float* __restrict__ msg) {
  __shared__ float Ss[16 * 128];
  __shared__ float Rs[16 * 128];
  const int row0 = blockIdx.x * 16;
  const int tid  = threadIdx.x;
  {
    const int r = tid >> 4;
    const int c = (tid & 15) * 8;
    *(float4*)(Ss + r * 128 + c)     = *(const float4*)(s_in + (size_t)(row0 + r) * 128 + c);
    *(float4*)(Ss + r * 128 + c + 4) = *(const float4*)(s_in + (size_t)(row0 + r) * 128 + c + 4);
  }
  __syncthreads();
  const int col0 = (tid >> 5) * 16;
  const int lane = tid & 31, l15 = lane & 15, half = lane >> 4;

  v8f az = load_c(xz + (size_t)row0 * 128, col0);
  v8f ar = load_c(xr + (size_t)row0 * 128, col0);
  az = gemm_tile(Ss, 128, Uz, 128, col0, az);
  ar = gemm_tile(Ss, 128, Ur, 128, col0, ar);

  v8f z, rg;
#pragma unroll
  for (int i = 0; i < 8; ++i) { z[i] = sigmoidf_(az[i]); rg[i] = sigmoidf_(ar[i]); }

#pragma unroll
  for (int i = 0; i < 8; ++i) {
    const int m = i + 8 * half, c = col0 + l15;
    Rs[m * 128 + c] = rg[i] * Ss[m * 128 + c];
  }
  __syncthreads();

  v8f ah = load_c(xh + (size_t)row0 * 128, col0);
  ah = gemm_tile(Rs, 128, Uh, 128, col0, ah);

#pragma unroll
  for (int i = 0; i < 8; ++i) {
    const int m = i + 8 * half, c = col0 + l15;
    const float sv = Ss[m * 128 + c];
    const float h  = tanhf(ah[i]);
    const float zi = z[i];
    msg[(size_t)(row0 + m) * 128 + c] = (1.0f - zi) * sv + zi * h;
  }
}

// scatter-add 128-float rows: out[dst[e]] += msg[src ? src[e] : e]
__global__ void k_scatter(const float* __restrict__ msg, const int* __restrict__ src,
                          const int* __restrict__ dst, float* __restrict__ out,
                          int nedges) {
  const long long gid = (long long)blockIdx.x * blockDim.x + threadIdx.x;
  const int e = (int)(gid >> 5);
  if (e >= nedges) return;
  const int q = (int)(gid & 31);
  const int s = src ? src[e] : e;
  const int d = dst[e];
  const float4 v = *(const float4*)(msg + (size_t)s * 128 + q * 4);
  float* o = out + (size_t)d * 128 + q * 4;
  atomicAdd(o + 0, v.x);
  atomicAdd(o + 1, v.y);
  atomicAdd(o + 2, v.z);
  atomicAdd(o + 3, v.w);
}

__global__ void k_gather_rows(const float* __restrict__ tbl, const int* __restrict__ idx,
                              float* __restrict__ out, int n) {
  const long long gid = (long long)blockIdx.x * blockDim.x + threadIdx.x;
  const int i = (int)(gid >> 5);
  if (i >= n) return;
  const int q = (int)(gid & 31);
  *(float4*)(out + (size_t)i * 128 + q * 4) =
      *(const float4*)(tbl + (size_t)idx[i] * 128 + q * 4);
}

__global__ void k_zero(float* __restrict__ p, long long n4) {
  const long long i = (long long)blockIdx.x * blockDim.x + threadIdx.x;
  if (i < n4) ((float4*)p)[i] = make_float4(0.f, 0.f, 0.f, 0.f);
}

__global__ void k_relu_copy(const float* __restrict__ in, float* __restrict__ out,
                            long long n4) {
  const long long i = (long long)blockIdx.x * blockDim.x + threadIdx.x;
  if (i >= n4) return;
  float4 v = ((const float4*)in)[i];
  v.x = fmaxf(v.x, 0.f); v.y = fmaxf(v.y, 0.f);
  v.z = fmaxf(v.z, 0.f); v.w = fmaxf(v.w, 0.f);
  ((float4*)out)[i] = v;
}

// first GRU step with s == 0: msg = sigmoid(xz) * tanh(xh)
__global__ void k_gru_first(const float* __restrict__ xz, const float* __restrict__ xh,
                            float* __restrict__ msg, long long n) {
  const long long i = (long long)blockIdx.x * blockDim.x + threadIdx.x;
  if (i >= n) return;
  msg[i] = sigmoidf_(xz[i]) * tanhf(xh[i]);
}

// -------------------- host --------------------
static inline int cdiv(long long a, long long b) { return (int)((a + b - 1) / b); }

extern "C" void kernel_launch(void* const* d_in, const int* in_sizes, int n_in,
                              void* d_out, int out_size, void* d_ws, size_t ws_size,
                              hipStream_t stream) {
  (void)in_sizes; (void)n_in; (void)out_size; (void)ws_size;
  const float* Gf    = (const float*)d_in[0];
  const float* Gef   = (const float*)d_in[1];
  const int*   Gsrc  = (const int*)d_in[2];
  const int*   Gdst  = (const int*)d_in[3];
  const int*   GLs   = (const int*)d_in[4];
  const int*   GLd   = (const int*)d_in[5];
  const int*   Tid   = (const int*)d_in[6];
  const int*   Tsrc  = (const int*)d_in[7];
  const int*   Tdst  = (const int*)d_in[8];
  const int*   TLs   = (const int*)d_in[9];
  const int*   TLd   = (const int*)d_in[10];
  const float* emb   = (const float*)d_in[11];
  const float* W1    = (const float*)d_in[12];
  const float* W2    = (const float*)d_in[13];
  const float* W3    = (const float*)d_in[14];
  const float* U1G   = (const float*)d_in[15];
  const float* U2G   = (const float*)d_in[16];
  const float* Wz    = (const float*)d_in[17];
  const float* Uz    = (const float*)d_in[18];
  const float* Wr    = (const float*)d_in[19];
  const float* Ur    = (const float*)d_in[20];
  const float* Wh    = (const float*)d_in[21];
  const float* Uh    = (const float*)d_in[22];
  const float* U1T   = (const float*)d_in[23];
  const float* U2T   = (const float*)d_in[24];
  float* out = (float*)d_out;

  const size_t EG128 = (size_t)EG * 128;
  const size_t ET128 = (size_t)ET * 128;
  const size_t NG128 = (size_t)NG * 128;
  const size_t NT128 = (size_t)NT * 128;

  float* W    = (float*)d_ws;
  float* bufA = W;               // base_G; later xz|xr|xh
  float* bufB = W + EG128;       // msgG;   later msgT|f_T|sT
  float* bufC = W + 2 * EG128;   // sum_msg; later sG|sumT

  // ============ molecule graph G ============
  float* baseG = bufA;
  float* msgG  = bufB;
  float* sumG  = bufC;

  k_base_G<<<EG / 16, 256, 0, stream>>>(Gf, Gef, Gsrc, W1, W2, baseG);
  // iteration 1: sum==0 -> msg = relu(base)
  k_relu_copy<<<cdiv(EG128 / 4, 256), 256, 0, stream>>>(baseG, msgG, (long long)(EG128 / 4));
  for (int it = 0; it < 3; ++it) {
    k_zero<<<cdiv(EG128 / 4, 256), 256, 0, stream>>>(sumG, (long long)(EG128 / 4));
    k_scatter<<<cdiv((long long)ELG * 32, 256), 256, 0, stream>>>(msgG, GLs, GLd, sumG, ELG);
    k_msg_update<<<EG / 16, 256, 0, stream>>>(sumG, baseG, W3, msgG);
  }
  float* sG = bufC;  // sum_msg space no longer needed as-is
  k_zero<<<cdiv(NG128 / 4, 256), 256, 0, stream>>>(sG, (long long)(NG128 / 4));
  k_scatter<<<cdiv((long long)EG * 32, 256), 256, 0, stream>>>(msgG, nullptr, Gdst, sG, EG);
  k_readout<64><<<NG / 16, 256, 0, stream>>>(Gf, sG, U1G, U2G, out);

  // ============ junction tree T ============
  float* xz   = bufA;
  float* xr   = bufA + ET128;
  float* xh   = bufA + 2 * ET128;
  float* msgT = bufB;
  float* fT   = bufB + ET128;
  float* sT   = bufB + ET128 + NT128;
  float* sumT = bufC + NG128;

  k_gather_rows<<<cdiv((long long)NT * 32, 256), 256, 0, stream>>>(emb, Tid, fT, NT);
  k_xzrh<<<ET / 16, 256, 0, stream>>>(fT, Tsrc, Wz, Wr, Wh, xz, xr, xh);
  // iteration 1: s==0 -> msg = sigmoid(xz) * tanh(xh)
  k_gru_first<<<cdiv((long long)ET128, 256), 256, 0, stream>>>(xz, xh, msgT, (long long)ET128);
  for (int it = 0; it < 3; ++it) {
    k_zero<<<cdiv(ET128 / 4, 256), 256, 0, stream>>>(sumT, (long long)(ET128 / 4));
    k_scatter<<<cdiv((long long)ELGT * 32, 256), 256, 0, stream>>>(msgT, TLs, TLd, sumT, ELGT);
    k_gru<<<ET / 16, 256, 0, stream>>>(sumT, xz, xr, xh, Uz, Ur, Uh, msgT);
  }
  k_zero<<<cdiv(NT128 / 4, 256), 256, 0, stream>>>(sT, (long long)(NT128 / 4));
  k_scatter<<<cdiv((long long)ET * 32, 256), 256, 0, stream>>>(msgT, nullptr, Tdst, sT, ET);
  k_readout<128><<<NT / 16, 256, 0, stream>>>(fT, sT, U1T, U2T, out + NG128);
}